// Region_Contra_Loss_90804198572475
// MI455X (gfx1250) — compile-verified
//
#include <hip/hip_runtime.h>

// ---------------------------------------------------------------------------
// Region contrastive loss (FID-style) for MI455X / gfx1250.
//   R=32 regions, C=128 channels, P=512*512 pixels.
//   Stage 1: per-region sums + second moments (X X^T) via V_WMMA_F32_16X16X4_F32
//   Stage 2: covariance assembly
//   Stage 3: sqrtm(sig_x + eps I) via Newton-Schulz (all 128^3 f32 GEMMs in LDS)
//   Stage 4: 32x64 pairwise Frechet distances (2 GEMMs + NS trace-sqrt per pair)
//   Stage 5: softmax-style loss reduction
// Data movement: matrices staged into LDS with GLOBAL_LOAD_ASYNC_TO_LDS_B128
// (ASYNCcnt / s_wait_asynccnt), mask stream prefetched with global_prefetch_b8.
// ---------------------------------------------------------------------------

#define RNUM 32
#define CNUM 128
#define PNUM (512 * 512)
#define EPSV 1e-6f
#define TEMPV 5.4f
#define MINPIX 10.0f
#define LP 132                 // LDS row pitch (dwords); 528 B rows stay 16B-aligned
#define MSZ (CNUM * LP)        // one 128x128 LDS matrix (padded)
#define NSLICE 8
#define NS_ITERS 14

typedef float v2f __attribute__((ext_vector_type(2)));
typedef float v8f __attribute__((ext_vector_type(8)));

__device__ __forceinline__ void atomAddG(float* p, float v) { unsafeAtomicAdd(p, v); }
__device__ __forceinline__ void atomAddL(float* p, float v) { atomicAdd(p, v); }

// ---- CDNA5 async global->LDS copy (no VGPR round trip, tracked by ASYNCcnt) ----
__device__ __forceinline__ void asyncLoadB128(void* ldsDst, const void* gSrc) {
  // Flat pointers into LDS carry the LDS byte offset in their low 32 bits.
  const unsigned ldsOff = (unsigned)(size_t)ldsDst;
  asm volatile("global_load_async_to_lds_b128 %0, %1, off"
               :: "v"(ldsOff), "v"(gSrc)
               : "memory");
}
__device__ __forceinline__ void asyncWait() {
  asm volatile("s_wait_asynccnt 0x0" ::: "memory");
}
// Stage a row-major 128x128 f32 matrix into LDS at pitch LP (16B chunks).
__device__ __forceinline__ void stageMatrixAsync(float* dstLds, const float* src,
                                                 int tid) {
  for (int q = tid; q < (CNUM * CNUM) / 4; q += 256) {
    const int row = q >> 5;   // 32 vec4 per row
    const int c4  = q & 31;
    asyncLoadB128(dstLds + row * LP + c4 * 4, src + row * CNUM + c4 * 4);
  }
}

// ---------------------------------------------------------------------------
// 128x128x128 f32 GEMM on LDS matrices (pitch LP), 256 threads = 8 waves.
// Wave w owns rows [16w,16w+16); 8 column tiles -> 64 accumulator VGPRs/wave.
// MODE 0: dst = A*B     MODE 1: dst = 0.5*A*B     MODE 2: dst = 3I - A*B
// ---------------------------------------------------------------------------
template <int MODE>
__device__ __forceinline__ void gemm128(float* dst, const float* A, const float* B) {
  const int tid  = threadIdx.x;
  const int lane = tid & 31;
  const int wv   = tid >> 5;
  const int hl   = lane >> 4;   // 0: K pair {0,1}, 1: K pair {2,3}
  const int mn   = lane & 15;
  v8f acc[8] = {};
  for (int k = 0; k < CNUM; k += 4) {
    const int kr = k + 2 * hl;
    v2f a;
    a.x = A[(16 * wv + mn) * LP + kr];
    a.y = A[(16 * wv + mn) * LP + kr + 1];
#pragma unroll
    for (int j = 0; j < 8; ++j) {
      v2f b;
      b.x = B[kr * LP + 16 * j + mn];
      b.y = B[(kr + 1) * LP + 16 * j + mn];
      acc[j] = __builtin_amdgcn_wmma_f32_16x16x4_f32(false, a, false, b,
                                                     (short)0, acc[j], false, false);
    }
  }
  __syncthreads();  // all operand reads done before dst (possibly aliased) is written
#pragma unroll
  for (int j = 0; j < 8; ++j) {
#pragma unroll
    for (int v = 0; v < 8; ++v) {
      const int row = 16 * wv + v + 8 * hl;   // C/D layout: VGPR v -> rows v, v+8
      const int col = 16 * j + mn;
      float r = acc[j][v];
      if (MODE == 1) r *= 0.5f;
      if (MODE == 2) r = ((row == col) ? 3.0f : 0.0f) - r;
      dst[row * LP + col] = r;
    }
  }
  __syncthreads();
}

// ---------------------------------------------------------------------------
// Stage 1: per-(region, map, slice) workgroup. Compact matching pixels into
// LDS, gather their 128-channel columns once, drain through WMMA outer-product
// accumulation (K = pixels, zero-padded to multiple of 4). Column sums and
// counts ride along. Accumulators flushed once via global_atomic_add_f32.
// ---------------------------------------------------------------------------
__global__ void __launch_bounds__(256)
k_stats(const float* __restrict__ xs, const float* __restrict__ xt,
        const int* __restrict__ mask,
        float* __restrict__ S2s, float* __restrict__ S2t,
        float* __restrict__ sums, float* __restrict__ sumt,
        float* __restrict__ cnt) {
  extern __shared__ float sm[];
  float* Xc   = sm;                       // [256][LP]  compacted pixel columns
  int*   lst  = (int*)(sm + 256 * LP);    // pixel index list
  int*   lcnt = lst + 256;

  const int r   = blockIdx.x;
  const int mp  = blockIdx.y;
  const int sl  = blockIdx.z;
  const int tid = threadIdx.x;
  const int lane = tid & 31;
  const int wv = tid >> 5, hl = lane >> 4, mn = lane & 15;

  const float* X = mp ? xt : xs;
  float* S2      = mp ? S2t : S2s;
  float* sum     = mp ? sumt : sums;

  const int PS   = PNUM / NSLICE;
  const int base = sl * PS;

  v8f acc[8] = {};

  for (int chunk = 0; chunk < PS; chunk += 256) {
    if (tid == 0) *lcnt = 0;
    __syncthreads();
    const int p = base + chunk + tid;
    if (chunk + 256 < PS) __builtin_prefetch(&mask[p + 256], 0, 0);
    if (mask[p] == r) {
      const int slot = atomicAdd(lcnt, 1);
      lst[slot] = p;
    }
    __syncthreads();
    const int n = *lcnt;
    __syncthreads();
    if (n == 0) continue;
    const int npad = (n + 3) & ~3;
    // gather columns: Xc[pixel][channel]
    for (int e = tid; e < npad * CNUM; e += 256) {
      const int kk = e >> 7, c = e & 127;
      float v = 0.0f;
      if (kk < n) v = X[(size_t)c * PNUM + lst[kk]];
      Xc[kk * LP + c] = v;
    }
    __syncthreads();
    if (tid < CNUM) {
      float s = 0.0f;
      for (int kk = 0; kk < n; ++kk) s += Xc[kk * LP + tid];
      atomAddG(&sum[r * CNUM + tid], s);
    }
    if (tid == 0 && mp == 0) atomAddG(&cnt[r], (float)n);
    // WMMA outer-product accumulation over pixel chunks of 4
    for (int k = 0; k < npad; k += 4) {
      const int kr = k + 2 * hl;
      v2f a;
      a.x = Xc[kr * LP + 16 * wv + mn];          // A[m][k] = X[chan m][pix k]
      a.y = Xc[(kr + 1) * LP + 16 * wv + mn];
#pragma unroll
      for (int j = 0; j < 8; ++j) {
        v2f b;
        b.x = Xc[kr * LP + 16 * j + mn];         // B[k][n] = X[chan n][pix k]
        b.y = Xc[(kr + 1) * LP + 16 * j + mn];
        acc[j] = __builtin_amdgcn_wmma_f32_16x16x4_f32(false, a, false, b,
                                                       (short)0, acc[j], false, false);
      }
    }
    __syncthreads();
  }
  // flush S2 partials
#pragma unroll
  for (int j = 0; j < 8; ++j) {
#pragma unroll
    for (int v = 0; v < 8; ++v) {
      const int row = 16 * wv + v + 8 * hl;
      const int col = 16 * j + mn;
      atomAddG(&S2[((size_t)r * CNUM + row) * CNUM + col], acc[j][v]);
    }
  }
}

// ---------------------------------------------------------------------------
// Stage 2: cov = (S2 - n*mu*mu^T)/max(n-1,1) in place; invalid regions -> I,
// mu -> 0 (matches reference masking).
// ---------------------------------------------------------------------------
__global__ void __launch_bounds__(256)
k_cov(const float* __restrict__ sums, const float* __restrict__ sumt,
      const float* __restrict__ cnt,
      float* __restrict__ sigx, float* __restrict__ sigy,
      float* __restrict__ mux, float* __restrict__ muy) {
  __shared__ float smu[CNUM];
  const int r = blockIdx.x, mp = blockIdx.y, tid = threadIdx.x;
  const float* sum = mp ? sumt : sums;
  float* sig = mp ? sigy : sigx;
  float* mu  = mp ? muy : mux;
  const float n = cnt[r];
  const bool valid = n >= MINPIX;
  const float nd = fmaxf(n, 1.0f);
  if (tid < CNUM) {
    const float m = sum[r * CNUM + tid] / nd;
    smu[tid] = m;
    mu[r * CNUM + tid] = valid ? m : 0.0f;
  }
  __syncthreads();
  const float sc = 1.0f / fmaxf(n - 1.0f, 1.0f);
  for (int e = tid; e < CNUM * CNUM; e += 256) {
    const int row = e >> 7, col = e & 127;
    float cv = (sig[(size_t)r * CNUM * CNUM + e] - n * smu[row] * smu[col]) * sc;
    if (!valid) cv = (row == col) ? 1.0f : 0.0f;
    sig[(size_t)r * CNUM * CNUM + e] = cv;
  }
}

// ---------------------------------------------------------------------------
// Stage 3: sA = sqrtm(sig_x + eps I) per region via Newton-Schulz, fully LDS-
// resident (4 matrices, 270 KB of the 320 KB WGP LDS). Input staged with
// async global->LDS copies.
// ---------------------------------------------------------------------------
__global__ void __launch_bounds__(256)
k_sqrtm(const float* __restrict__ sigx, float* __restrict__ sAo) {
  extern __shared__ float sm[];
  float* B0 = sm;
  float* B1 = sm + MSZ;
  float* B2 = sm + 2 * MSZ;
  float* B3 = sm + 3 * MSZ;
  float* ctl = sm + 4 * MSZ;
  const int i = blockIdx.x, tid = threadIdx.x;

  if (tid == 0) ctl[0] = 0.0f;
  stageMatrixAsync(B2, sigx + (size_t)i * CNUM * CNUM, tid);
  asyncWait();
  __syncthreads();
  if (tid < CNUM) B2[tid * LP + tid] += EPSV;
  __syncthreads();
  if (tid < CNUM) atomAddL(&ctl[0], B2[tid * LP + tid]);
  __syncthreads();
  const float t = fmaxf(ctl[0], 1e-20f);
  for (int e = tid; e < CNUM * CNUM; e += 256) {
    const int r = e >> 7, c = e & 127;
    B3[r * LP + c] = B2[r * LP + c] / t;            // Y0 = A/t
    B1[r * LP + c] = (r == c) ? 1.0f : 0.0f;        // Z0 = I
  }
  __syncthreads();
  float* Y = B3; float* Z = B1; float* T = B2; float* W = B0;
  for (int it = 0; it < NS_ITERS; ++it) {
    gemm128<2>(T, Z, Y);   // T = 3I - Z*Y
    gemm128<1>(W, Y, T);   // Ynew = 0.5*Y*T
    gemm128<1>(Y, T, Z);   // Znew = 0.5*T*Z  (into old Y slot)
    float* oy = Y; Y = W; W = Z; Z = oy;
  }
  const float sq = sqrtf(t);
  for (int e = tid; e < CNUM * CNUM; e += 256) {
    const int r = e >> 7, c = e & 127;
    sAo[(size_t)i * CNUM * CNUM + e] = sq * Y[r * LP + c];
  }
}

// ---------------------------------------------------------------------------
// Stage 4: D[i][j] = |mu_i-mu_j|^2 + tr(s1) + tr(s2) - 2*tr(sqrtm(M)),
// M = sA_i (sig_b_j + eps I) sA_i, symmetrized; trace-sqrt via Newton-Schulz.
// Matrix staging via async global->LDS; scalar reductions overlap the DMA.
// ---------------------------------------------------------------------------
__global__ void __launch_bounds__(256)
k_pairs(const float* __restrict__ sA,
        const float* __restrict__ sigx, const float* __restrict__ sigy,
        const float* __restrict__ mux, const float* __restrict__ muy,
        float* __restrict__ Dm) {
  extern __shared__ float sm[];
  float* B0 = sm;
  float* B1 = sm + MSZ;
  float* B2 = sm + 2 * MSZ;
  float* B3 = sm + 3 * MSZ;
  float* ctl = sm + 4 * MSZ;
  const int i = blockIdx.y, j = blockIdx.x, tid = threadIdx.x;
  const float* sb = (j < RNUM) ? (sigx + (size_t)j * CNUM * CNUM)
                               : (sigy + (size_t)(j - RNUM) * CNUM * CNUM);
  const float* mb = (j < RNUM) ? (mux + j * CNUM) : (muy + (j - RNUM) * CNUM);

  if (tid < 8) ctl[tid] = 0.0f;
  __syncthreads();
  stageMatrixAsync(B0, sA + (size_t)i * CNUM * CNUM, tid);
  stageMatrixAsync(B1, sb, tid);
  // small reductions overlap the async transfers
  if (tid < CNUM) {
    const float d = mux[i * CNUM + tid] - mb[tid];
    atomAddL(&ctl[1], d * d);                                      // diff.diff
    atomAddL(&ctl[2], sigx[(size_t)i * CNUM * CNUM + tid * 129]);  // tr(s1)
    atomAddL(&ctl[3], sb[tid * 129]);                              // tr(s2)
  }
  asyncWait();
  __syncthreads();
  if (tid < CNUM) B1[tid * LP + tid] += EPSV;
  __syncthreads();
  gemm128<0>(B2, B0, B1);   // sA * (s2 + eps I)
  gemm128<0>(B3, B2, B0);   // M = sA * (s2+eps I) * sA
  for (int e = tid; e < CNUM * CNUM; e += 256) {                // 0.5*(M+M^T)
    const int r = e >> 7, c = e & 127;
    B2[r * LP + c] = 0.5f * (B3[r * LP + c] + B3[c * LP + r]);
  }
  __syncthreads();
  if (tid < CNUM) atomAddL(&ctl[0], B2[tid * LP + tid]);
  __syncthreads();
  const float t = fmaxf(ctl[0], 1e-20f);
  for (int e = tid; e < CNUM * CNUM; e += 256) {
    const int r = e >> 7, c = e & 127;
    B3[r * LP + c] = B2[r * LP + c] / t;            // Y0 = M/t
    B1[r * LP + c] = (r == c) ? 1.0f : 0.0f;        // Z0 = I
  }
  __syncthreads();
  float* Y = B3; float* Z = B1; float* T = B2; float* W = B0;
  for (int it = 0; it < NS_ITERS; ++it) {
    gemm128<2>(T, Z, Y);
    gemm128<1>(W, Y, T);
    gemm128<1>(Y, T, Z);
    float* oy = Y; Y = W; W = Z; Z = oy;
  }
  if (tid < CNUM) atomAddL(&ctl[4], Y[tid * LP + tid]);
  __syncthreads();
  if (tid == 0) {
    const float tr_cov = sqrtf(t) * ctl[4];
    Dm[i * 2 * RNUM + j] = ctl[1] + ctl[2] + ctl[3] - 2.0f * tr_cov;
  }
}

// ---------------------------------------------------------------------------
// Stage 5: contrastive reduction -> scalar loss.
// ---------------------------------------------------------------------------
__global__ void k_loss(const float* __restrict__ Dm, const float* __restrict__ cnt,
                       float* __restrict__ out) {
  __shared__ float sval[RNUM];
  __shared__ float spart[RNUM];
  __shared__ float snum;
  const int tid = threadIdx.x;
  if (tid < RNUM) sval[tid] = (cnt[tid] >= MINPIX) ? 1.0f : 0.0f;
  __syncthreads();
  if (tid == 0) {
    float s = 0.0f;
    for (int k = 0; k < RNUM; ++k) s += sval[k];
    snum = s;
  }
  __syncthreads();
  const float num_r = snum;
  if (tid < RNUM) {
    float s = 0.0f;
    for (int j = 0; j < 2 * RNUM; ++j)
      if (sval[j & (RNUM - 1)] != 0.0f) s += expf(-Dm[tid * 2 * RNUM + j] / TEMPV);
    const float exp_neg = s / (2.0f * num_r);
    const float lr = -logf(1.0f / (exp_neg + EPSV) + EPSV);
    spart[tid] = (sval[tid] != 0.0f) ? lr : 0.0f;
  }
  __syncthreads();
  if (tid == 0) {
    float s = 0.0f;
    for (int k = 0; k < RNUM; ++k) s += spart[k];
    out[0] = s / num_r;
  }
}

// ---------------------------------------------------------------------------
extern "C" void kernel_launch(void* const* d_in, const int* in_sizes, int n_in,
                              void* d_out, int out_size, void* d_ws, size_t ws_size,
                              hipStream_t stream) {
  (void)in_sizes; (void)n_in; (void)out_size; (void)ws_size;
  const float* map_s = (const float*)d_in[0];
  const float* map_t = (const float*)d_in[1];
  const int*   mask  = (const int*)d_in[2];
  float* out = (float*)d_out;
  float* ws  = (float*)d_ws;

  float* cnt  = ws;                       // 32
  float* sums = cnt + RNUM;               // 32*128
  float* sumt = sums + RNUM * CNUM;       // 32*128
  float* sigx = sumt + RNUM * CNUM;       // 32*128*128 (S2 -> cov in place)
  float* sigy = sigx + RNUM * CNUM * CNUM;
  float* mux  = sigy + RNUM * CNUM * CNUM;
  float* muy  = mux + RNUM * CNUM;
  float* sA   = muy + RNUM * CNUM;        // 32*128*128
  float* Dm   = sA + RNUM * CNUM * CNUM;  // 32*64

  const size_t zero_bytes =
      (size_t)(RNUM + 2 * RNUM * CNUM + 2 * RNUM * CNUM * CNUM) * sizeof(float);
  hipMemsetAsync(ws, 0, zero_bytes, stream);

  const size_t statsLds = (size_t)(256 * LP) * sizeof(float) + 256 * sizeof(int) + 16;
  k_stats<<<dim3(RNUM, 2, NSLICE), 256, statsLds, stream>>>(
      map_s, map_t, mask, sigx, sigy, sums, sumt, cnt);

  k_cov<<<dim3(RNUM, 2), 256, 0, stream>>>(sums, sumt, cnt, sigx, sigy, mux, muy);

  const size_t matLds = (size_t)(4 * MSZ + 16) * sizeof(float);
  k_sqrtm<<<RNUM, 256, matLds, stream>>>(sigx, sA);
  k_pairs<<<dim3(2 * RNUM, RNUM), 256, matLds, stream>>>(sA, sigx, sigy, mux, muy, Dm);
  k_loss<<<1, 32, 0, stream>>>(Dm, cnt, out);
}